// guided_CAM_Module_50646254355143
// MI455X (gfx1250) — compile-verified
//
#include <hip/hip_runtime.h>

typedef float v2f __attribute__((ext_vector_type(2)));
typedef float v8f __attribute__((ext_vector_type(8)));

#define BB 8
#define CC 320
#define QQ 64
#define SS 16384   // H*W = 128*128

// ---------------------------------------------------------------------------
// Kernel 1: energy[b,q,c] = sum_s guide[b,q,s] * x[b,c,s]
// One 16(q) x 16(c) tile per 256-thread block; 8 waves split K = S into
// 2048-element slices, each wave runs 512 V_WMMA_F32_16X16X4_F32 ops,
// then partial tiles are reduced through LDS.
// A-fragment (lane l): { guide[q0+(l&15)][k+2*(l>>4)], ...+1 }  (contig float2)
// B-fragment (lane l): { x[c0+(l&15)][k+2*(l>>4)],     ...+1 }  (contig float2)
// ---------------------------------------------------------------------------
__global__ __launch_bounds__(256)
void energy_kernel(const float* __restrict__ x, const float* __restrict__ guide,
                   float* __restrict__ energy) {
  const int c0   = blockIdx.x * 16;
  const int q0   = blockIdx.y * 16;
  const int b    = blockIdx.z;
  const int tid  = threadIdx.x;
  const int wave = tid >> 5;
  const int lane = tid & 31;
  const int hi   = lane >> 4;
  const int m    = lane & 15;

  const int KSLICE = SS / 8; // 2048 per wave
  const float* g  = guide + (size_t)(b * QQ + q0 + m) * SS + wave * KSLICE + 2 * hi;
  const float* xp = x     + (size_t)(b * CC + c0 + m) * SS + wave * KSLICE + 2 * hi;

  v8f acc = {};
  #pragma unroll 4
  for (int k = 0; k < KSLICE; k += 4) {
    v2f a  = *(const v2f*)(g  + k);
    v2f bb = *(const v2f*)(xp + k);
    acc = __builtin_amdgcn_wmma_f32_16x16x4_f32(false, a, false, bb,
                                                (short)0, acc, false, false);
  }

  // Cross-wave reduction of the 8 partial 16x16 tiles.
  __shared__ float part[8][8][32];
  #pragma unroll
  for (int v = 0; v < 8; ++v) part[wave][v][lane] = acc[v];
  __syncthreads();

  const int v = tid >> 5;   // which of the 8 accumulator VGPRs
  const int l = tid & 31;   // which lane
  float s = 0.f;
  #pragma unroll
  for (int w = 0; w < 8; ++w) s += part[w][v][l];

  // C/D layout: element (v, l) -> m = v + 8*(l>>4), n = l&15
  const int mo = v + 8 * (l >> 4);
  const int no = l & 15;
  energy[(size_t)(b * QQ + q0 + mo) * CC + c0 + no] = s;
}

// ---------------------------------------------------------------------------
// Kernel 2: in-place softmax. softmax(rowmax - e) == exp(min_c e - e)/sum.
// One block per (b,q) row of 320 channels.
// ---------------------------------------------------------------------------
__global__ __launch_bounds__(256)
void softmax_kernel(float* __restrict__ e) {
  float* row = e + (size_t)blockIdx.x * CC;
  const int t = threadIdx.x;
  __shared__ float red[256];

  float v0 = row[t];
  float v1 = (t + 256 < CC) ? row[t + 256] : 3.4e38f;
  red[t] = fminf(v0, v1);
  __syncthreads();
  for (int o = 128; o > 0; o >>= 1) {
    if (t < o) red[t] = fminf(red[t], red[t + o]);
    __syncthreads();
  }
  const float mn = red[0];
  __syncthreads();

  float e0 = __expf(mn - v0);
  float e1 = (t + 256 < CC) ? __expf(mn - v1) : 0.f;
  red[t] = e0 + e1;
  __syncthreads();
  for (int o = 128; o > 0; o >>= 1) {
    if (t < o) red[t] += red[t + o];
    __syncthreads();
  }
  const float inv = 1.0f / red[0];
  row[t] = e0 * inv;
  if (t + 256 < CC) row[t + 256] = e1 * inv;
}

// ---------------------------------------------------------------------------
// Kernel 3: out[b,q,s] = sum_c att[b,q,c] * x[b,c,s]
// Each wave: one 16(q) x 64(s) strip = 4 N-tiles sharing the A fragment.
// K-loop over C=320 in steps of 4 (80 WMMAs per tile).
// B-fragment loads are coalesced along s (row-major in memory).
// ---------------------------------------------------------------------------
__global__ __launch_bounds__(256)
void out_kernel(const float* __restrict__ x, const float* __restrict__ att,
                float* __restrict__ out) {
  const int sb   = blockIdx.x;   // 32 blocks of 512 s
  const int qt   = blockIdx.y;   // 4
  const int b    = blockIdx.z;   // 8
  const int tid  = threadIdx.x;
  const int wave = tid >> 5;
  const int lane = tid & 31;
  const int hi   = lane >> 4;
  const int mn   = lane & 15;    // M for the A fragment, N for B/D

  const int q0 = qt * 16;
  const int s0 = sb * 512 + wave * 64;

  const float* arow = att + (size_t)(b * QQ + q0 + mn) * CC + 2 * hi;
  const float* xb   = x + (size_t)b * CC * SS + (size_t)(2 * hi) * SS + s0 + mn;

  v8f acc0 = {}, acc1 = {}, acc2 = {}, acc3 = {};
  #pragma unroll 2
  for (int k = 0; k < CC; k += 4) {
    v2f a = *(const v2f*)(arow + k);
    const float* xr = xb + (size_t)k * SS;
    v2f b0 = { xr[0],  xr[SS]      };
    v2f b1 = { xr[16], xr[SS + 16] };
    v2f b2 = { xr[32], xr[SS + 32] };
    v2f b3 = { xr[48], xr[SS + 48] };
    acc0 = __builtin_amdgcn_wmma_f32_16x16x4_f32(false, a, false, b0, (short)0, acc0, false, false);
    acc1 = __builtin_amdgcn_wmma_f32_16x16x4_f32(false, a, false, b1, (short)0, acc1, false, false);
    acc2 = __builtin_amdgcn_wmma_f32_16x16x4_f32(false, a, false, b2, (short)0, acc2, false, false);
    acc3 = __builtin_amdgcn_wmma_f32_16x16x4_f32(false, a, false, b3, (short)0, acc3, false, false);
  }

  float* orow = out + (size_t)(b * QQ + q0) * SS + s0;
  #pragma unroll
  for (int v = 0; v < 8; ++v) {
    const int mo = v + 8 * hi;
    float* dst = orow + (size_t)mo * SS + mn;
    dst[0]  = acc0[v];
    dst[16] = acc1[v];
    dst[32] = acc2[v];
    dst[48] = acc3[v];
  }
}

// ---------------------------------------------------------------------------
extern "C" void kernel_launch(void* const* d_in, const int* in_sizes, int n_in,
                              void* d_out, int out_size, void* d_ws, size_t ws_size,
                              hipStream_t stream) {
  const float* x     = (const float*)d_in[0];  // (B, C, H, W)
  const float* guide = (const float*)d_in[1];  // (B, Q, H, W)
  float* out    = (float*)d_out;               // (B, Q, H, W)
  float* energy = (float*)d_ws;                // (B, Q, C) = 655 KB, softmaxed in place

  dim3 g1(CC / 16, QQ / 16, BB);               // 20 x 4 x 8 = 640 blocks
  energy_kernel<<<g1, 256, 0, stream>>>(x, guide, energy);

  softmax_kernel<<<BB * QQ, 256, 0, stream>>>(energy);

  dim3 g3(SS / 512, QQ / 16, BB);              // 32 x 4 x 8 = 1024 blocks
  out_kernel<<<g3, 256, 0, stream>>>(x, energy, out);
}